// MedianFilter_27161373180312
// MI455X (gfx1250) — compile-verified
//
#include <hip/hip_runtime.h>
#include <hip/hip_bf16.h>

// 3x3 median filter, replicate border (cv2.medianBlur semantics).
// Memory-bound: 256 MiB min traffic -> ~11 us floor @ 23.3 TB/s HBM.
// Strategy: register-rolling row window, 8 cols x 16 rows per thread,
// exact Smith median-of-9 network on v_med3_num_f32 / v_min3 / v_max3.
// One 256-thread block (8 wave32) spans the full 2048-px row width.

#define MF_VEC  8   // columns per thread (two float4)
#define MF_ROWS 16  // output rows per thread
#define MF_TPB  256 // threads per block = 8 wave32

__device__ __forceinline__ float med3f(float a, float b, float c) {
    return __builtin_amdgcn_fmed3f(a, b, c);
}
__device__ __forceinline__ float min3f(float a, float b, float c) {
    return fminf(fminf(a, b), c);   // backend fuses to v_min3_num_f32
}
__device__ __forceinline__ float max3f(float a, float b, float c) {
    return fmaxf(fmaxf(a, b), c);   // backend fuses to v_max3_num_f32
}

__global__ __launch_bounds__(MF_TPB)
void MedianFilter3x3_kernel(const float* __restrict__ in,
                            float* __restrict__ out,
                            int H, int W) {
    const int x0 = (blockIdx.x * blockDim.x + threadIdx.x) * MF_VEC;
    const int y0 = blockIdx.y * MF_ROWS;
    const long long plane = (long long)H * (long long)W;
    const float* __restrict__ base  = in  + (long long)blockIdx.z * plane;
    float* __restrict__       obase = out + (long long)blockIdx.z * plane;

    const int xl = (x0 > 0) ? (x0 - 1) : 0;                      // replicate left
    const int xr = (x0 + MF_VEC < W) ? (x0 + MF_VEC) : (W - 1);  // replicate right

    // Load one horizontally-extended row: [left, v0..v7, right]
    auto loadrow = [&](int r, float v[MF_VEC + 2]) {
        r = (r < 0) ? 0 : ((r > H - 1) ? (H - 1) : r);           // replicate top/bottom
        const float* rp = base + (long long)r * W;
        const float4 m0 = *reinterpret_cast<const float4*>(rp + x0);
        const float4 m1 = *reinterpret_cast<const float4*>(rp + x0 + 4);
        v[0] = rp[xl];
        v[1] = m0.x; v[2] = m0.y; v[3] = m0.z; v[4] = m0.w;
        v[5] = m1.x; v[6] = m1.y; v[7] = m1.z; v[8] = m1.w;
        v[9] = rp[xr];
    };

    float a[MF_VEC + 2], b[MF_VEC + 2], c[MF_VEC + 2];
    loadrow(y0 - 1, a);
    loadrow(y0,     b);

    #pragma unroll
    for (int i = 0; i < MF_ROWS; ++i) {
        const int y = y0 + i;
        loadrow(y + 1, c);

        // Prefetch 3 rows ahead (gfx1250 global_prefetch_b8); 256 lanes x 32B
        // spans the whole upcoming row.
        {
            int rp2 = y + 3; rp2 = (rp2 > H - 1) ? (H - 1) : rp2;
            __builtin_prefetch(base + (long long)rp2 * W + x0, 0, 1);
        }

        // Vertical sort of each of the 10 columns: lo/mid/hi (3 instrs/col)
        float lo[MF_VEC + 2], mi[MF_VEC + 2], hi[MF_VEC + 2];
        #pragma unroll
        for (int k = 0; k < MF_VEC + 2; ++k) {
            lo[k] = min3f(a[k], b[k], c[k]);
            hi[k] = max3f(a[k], b[k], c[k]);
            mi[k] = med3f(a[k], b[k], c[k]);
        }

        // Horizontal combine: median9 = med3(max3(lo), med3(mi), min3(hi))
        float o[MF_VEC];
        #pragma unroll
        for (int j = 0; j < MF_VEC; ++j) {
            const float mxlo = max3f(lo[j], lo[j + 1], lo[j + 2]);
            const float mdmi = med3f(mi[j], mi[j + 1], mi[j + 2]);
            const float mnhi = min3f(hi[j], hi[j + 1], hi[j + 2]);
            o[j] = med3f(mxlo, mdmi, mnhi);
        }

        float* wp = obase + (long long)y * W + x0;
        *reinterpret_cast<float4*>(wp)     = make_float4(o[0], o[1], o[2], o[3]);
        *reinterpret_cast<float4*>(wp + 4) = make_float4(o[4], o[5], o[6], o[7]);

        // Roll the window down one row (register renaming after full unroll)
        #pragma unroll
        for (int k = 0; k < MF_VEC + 2; ++k) { a[k] = b[k]; b[k] = c[k]; }
    }
}

extern "C" void kernel_launch(void* const* d_in, const int* in_sizes, int n_in,
                              void* d_out, int out_size, void* d_ws, size_t ws_size,
                              hipStream_t stream) {
    (void)in_sizes; (void)n_in; (void)out_size; (void)d_ws; (void)ws_size;
    const float* x = (const float*)d_in[0];
    float* out = (float*)d_out;

    const int Bn = 8, H = 2048, W = 2048;
    dim3 block(MF_TPB, 1, 1);
    dim3 grid(W / (MF_VEC * MF_TPB),  // 2048/2048 = 1 (block spans full width)
              H / MF_ROWS,            // 2048/16   = 128
              Bn);                    // 8
    MedianFilter3x3_kernel<<<grid, block, 0, stream>>>(x, out, H, W);
}